// LocationExpertRouter_53936199303557
// MI455X (gfx1250) — compile-verified
//
#include <hip/hip_runtime.h>
#include <hip/hip_bf16.h>

// ---------------------------------------------------------------------------
// LocationExpertRouter: out[b,:] = x[b,:] @ W[ptr[b]%8] + bias[ptr[b]%8]
//   x:    [2048, 1024]  f32
//   ptr:  [2048]        i64
//   W:    [8, 1024, 32000] f32   (row k of W[e] contiguous over vocab)
//   bias: [8, 32000]    f32
//   out:  [2048, 32000] f32
// ---------------------------------------------------------------------------

#define BATCH     2048
#define DMODEL    1024
#define VOCAB     32000
#define NEXPERTS  8
#define ROWS      64                          // rows per super-tile (4 M-tiles)
#define MAX_ST    (BATCH / ROWS + NEXPERTS)   // 40 worst-case super-tiles
#define KCHUNK    32
#define NB        128                         // columns per block (4 waves * 32)
#define ASTRIDE   36                          // LDS row stride (floats): 144B, 16B-aligned
#define MT        4                           // M-tiles per wave
#define NT        2                           // N-tiles per wave

typedef __attribute__((ext_vector_type(2))) float v2f;
typedef __attribute__((ext_vector_type(4))) float v4f;
typedef __attribute__((ext_vector_type(8))) float v8f;

// ws layout (ints): [0] = total super-tiles
//                   [1 .. 1+MAX_ST)              = tile_expert[]
//                   [1+MAX_ST .. +MAX_ST*ROWS)   = perm[] (token id or -1)

__global__ __launch_bounds__(256)
void router_build_kernel(const long long* __restrict__ ptr, int* __restrict__ ws) {
    __shared__ int eidx[BATCH];
    __shared__ int cnt[NEXPERTS];
    __shared__ int tbase[NEXPERTS];
    const int tid = threadIdx.x;

    for (int i = tid; i < BATCH; i += 256)
        eidx[i] = (int)(ptr[i] & (NEXPERTS - 1));   // ptr >= 0, so %8 == &7
    __syncthreads();

    if (tid < NEXPERTS) {
        int c = 0;
        for (int i = 0; i < BATCH; ++i)
            if (eidx[i] == tid) ++c;
        cnt[tid] = c;
    }
    __syncthreads();

    if (tid == 0) {
        int tb = 0;
        for (int e = 0; e < NEXPERTS; ++e) {
            tbase[e] = tb;
            tb += (cnt[e] + ROWS - 1) / ROWS;
        }
        ws[0] = tb;   // total super-tiles actually used
    }
    __syncthreads();

    int* tile_expert = ws + 1;
    int* perm        = ws + 1 + MAX_ST;

    if (tid < NEXPERTS) {
        const int e     = tid;
        const int base  = tbase[e] * ROWS;
        const int ntile = (cnt[e] + ROWS - 1) / ROWS;
        int j = 0;
        for (int i = 0; i < BATCH; ++i)
            if (eidx[i] == e) perm[base + (j++)] = i;      // stable order
        for (; j < ntile * ROWS; ++j) perm[base + j] = -1; // padding rows
        for (int t = 0; t < ntile; ++t) tile_expert[tbase[e] + t] = e;
    }
}

// Grid: x = super-tile (fastest => all experts co-sweep the same 4MB W slab
//           through the 192MB L2; W streams from HBM once),
//       y = 128-wide vocab chunk (250).
// Block: 128 threads = 4 waves. Wave w owns a 64x32 output slab:
//        4 M-tiles (rows 0-15/16-31/32-47/48-63) x 2 N-tiles (cols +0/+16).
//        Each B fragment feeds 4 WMMAs -> 0.5 global dword loads per WMMA.
__global__ __launch_bounds__(128)
void moe_gemm_kernel(const float* __restrict__ x,
                     const float* __restrict__ W,
                     const float* __restrict__ bias,
                     float* __restrict__ out,
                     const int* __restrict__ ws) {
    const int total = ws[0];
    const int tile  = blockIdx.x;
    if (tile >= total) return;

    const int* tile_expert = ws + 1;
    const int* perm        = ws + 1 + MAX_ST;
    const int  e           = tile_expert[tile];

    __shared__ int tok[ROWS];
    __shared__ __align__(16) float As[ROWS * ASTRIDE];   // 64 x 36 floats (9 KB)

    const int tid  = threadIdx.x;
    const int wave = tid >> 5;
    const int lane = tid & 31;
    const int half = lane >> 4;   // 0: lanes 0-15, 1: lanes 16-31
    const int l15  = lane & 15;

    for (int i = tid; i < ROWS; i += 128) tok[i] = perm[tile * ROWS + i];
    __syncthreads();

    const int    n0 = blockIdx.y * NB + wave * 32;
    const float* Wb = W + (size_t)e * ((size_t)DMODEL * VOCAB);

    v8f acc[MT][NT];
#pragma unroll
    for (int mt = 0; mt < MT; ++mt)
#pragma unroll
        for (int nt = 0; nt < NT; ++nt)
            acc[mt][nt] = (v8f){};

    for (int k0 = 0; k0 < DMODEL; k0 += KCHUNK) {
        // --- stage A (64 rows x 32 K) into LDS, vectorized b128 ---
        __syncthreads();
#pragma unroll
        for (int i = 0; i < 4; ++i) {                 // 512 float4's / 128 threads
            const int idx = tid + i * 128;
            const int r   = idx >> 3;                 // row 0..63
            const int c4  = (idx & 7) * 4;            // col 0,4,...,28
            const int t   = tok[r];
            v4f v = {};
            if (t >= 0)
                v = *(const v4f*)(x + (size_t)t * DMODEL + k0 + c4);
            *(v4f*)(As + r * ASTRIDE + c4) = v;
        }
        __syncthreads();

#pragma unroll
        for (int ks = 0; ks < KCHUNK; ks += 4) {
            // A frags (16x4 f32): lanes 0-15 hold K=ks,ks+1; lanes 16-31 K=ks+2,ks+3
            v2f a[MT];
#pragma unroll
            for (int mt = 0; mt < MT; ++mt) {
                const int r = l15 + 16 * mt;
                a[mt].x = As[r * ASTRIDE + ks + half * 2 + 0];
                a[mt].y = As[r * ASTRIDE + ks + half * 2 + 1];
            }

            // B frags (4x16 f32), coalesced dword loads; reused by all 4 M-tiles.
            const float* Bp = Wb + (size_t)(k0 + ks) * VOCAB + n0 + l15;
            v2f b[NT];
#pragma unroll
            for (int nt = 0; nt < NT; ++nt) {
                b[nt].x = Bp[(size_t)(half * 2 + 0) * VOCAB + 16 * nt];
                b[nt].y = Bp[(size_t)(half * 2 + 1) * VOCAB + 16 * nt];
            }

#pragma unroll
            for (int mt = 0; mt < MT; ++mt)
#pragma unroll
                for (int nt = 0; nt < NT; ++nt)
                    acc[mt][nt] = __builtin_amdgcn_wmma_f32_16x16x4_f32(
                        false, a[mt], false, b[nt], (short)0, acc[mt][nt],
                        false, false);
        }
    }

    // C/D layout: VGPR r -> M = r (lanes 0-15) / M = r+8 (lanes 16-31), N = lane&15.
    const int   n   = n0 + l15;
    const float bv0 = bias[e * VOCAB + n];
    const float bv1 = bias[e * VOCAB + n + 16];

#pragma unroll
    for (int mt = 0; mt < MT; ++mt) {
#pragma unroll
        for (int r = 0; r < 8; ++r) {
            const int m = r + half * 8 + 16 * mt;
            const int t = tok[m];
            if (t >= 0) {
                out[(size_t)t * VOCAB + n]      = acc[mt][0][r] + bv0;
                out[(size_t)t * VOCAB + n + 16] = acc[mt][1][r] + bv1;
            }
        }
    }
}

extern "C" void kernel_launch(void* const* d_in, const int* in_sizes, int n_in,
                              void* d_out, int out_size, void* d_ws, size_t ws_size,
                              hipStream_t stream) {
    const float*     x    = (const float*)d_in[0];
    const long long* ptr  = (const long long*)d_in[1];
    const float*     W    = (const float*)d_in[2];
    const float*     bias = (const float*)d_in[3];
    float*           out  = (float*)d_out;
    int*             ws   = (int*)d_ws;

    router_build_kernel<<<1, 256, 0, stream>>>(ptr, ws);

    dim3 grid(MAX_ST, VOCAB / NB);   // (40, 250)
    moe_gemm_kernel<<<grid, 128, 0, stream>>>(x, W, bias, out, ws);
}